// GCN_32607391711761
// MI455X (gfx1250) — compile-verified
//
#include <hip/hip_runtime.h>
#include <hip/hip_bf16.h>
#include <math.h>

typedef __attribute__((ext_vector_type(2))) float v2f;
typedef __attribute__((ext_vector_type(8))) float v8f;

#define IN_DIM 128
#define HID    64

// ---------------- degree / normalization ----------------

__global__ __launch_bounds__(256) void gcn_deg_init(float* deg, int n) {
  int i = blockIdx.x * blockDim.x + threadIdx.x;
  if (i < n) deg[i] = 1.0f;  // self-loop contributes 1 to every node's degree
}

__global__ __launch_bounds__(256) void gcn_deg_count(const long long* __restrict__ ei,
                                                     float* deg, int e) {
  int i = blockIdx.x * blockDim.x + threadIdx.x;
  if (i < e) atomicAdd(&deg[(int)ei[e + i]], 1.0f);  // col = targets = ei[1]
}

__global__ __launch_bounds__(256) void gcn_dis(float* deg, int n) {
  int i = blockIdx.x * blockDim.x + threadIdx.x;
  if (i < n) {
    float d = deg[i];
    deg[i] = (d > 0.0f) ? rsqrtf(d) : 0.0f;
  }
}

// norm[e] for e < E: dis[row]*dis[col]; for e in [E, E+N): self-loop dis[i]^2
__global__ __launch_bounds__(256) void gcn_norm(const long long* __restrict__ ei,
                                                const float* __restrict__ dis,
                                                float* __restrict__ norm, int e, int n) {
  int i = blockIdx.x * blockDim.x + threadIdx.x;
  if (i < e) {
    norm[i] = dis[(int)ei[i]] * dis[(int)ei[e + i]];
  } else if (i < e + n) {
    float d = dis[i - e];
    norm[i] = d * d;
  }
}

// ---------------- GEMM1: H = X(Nx128) @ W1(128x64) via V_WMMA_F32_16X16X4_F32 ----

__global__ __launch_bounds__(256) void gcn_gemm1_wmma(const float* __restrict__ X,
                                                      const float* __restrict__ W1,
                                                      float* __restrict__ H, int n) {
  const int wave  = threadIdx.x >> 5;          // 8 waves per block
  const int lane  = threadIdx.x & 31;
  const int mtile = blockIdx.x * 2 + (wave >> 2);
  const int ntile = wave & 3;                  // 4 tiles cover HID=64
  const int lhalf = lane >> 4;                 // 0 for lanes 0-15, 1 for 16-31
  const int l16   = lane & 15;

  int m = mtile * 16 + l16;
  const int mclamp = (m < n) ? m : (n - 1);    // clamp loads; keep EXEC all-ones
  const int ncol   = ntile * 16 + l16;
  const int khalf  = lhalf * 2;                // K offset 0 or 2 within the x4 step

  v8f c = {};
  const float* xrow = X + (size_t)mclamp * IN_DIM;
  for (int k = 0; k < IN_DIM; k += 4) {
    v2f a, b;
    // A 16x4 f32: lane M = l16; VGPR0 -> K = khalf, VGPR1 -> K = khalf+1
    const float* ap = xrow + k + khalf;
    a.x = ap[0];
    a.y = ap[1];
    // B 4x16 f32 (row-striped across lanes): VGPR0 -> row K = khalf, VGPR1 -> K = khalf+1
    const float* bp = W1 + (size_t)(k + khalf) * HID + ncol;
    b.x = bp[0];
    b.y = bp[HID];
    c = __builtin_amdgcn_wmma_f32_16x16x4_f32(false, a, false, b, (short)0, c,
                                              false, false);
  }
  // D 16x16 f32: VGPR r -> row M = r + 8*lhalf, col = l16
  const int mbase = mtile * 16 + lhalf * 8;
#pragma unroll
  for (int r = 0; r < 8; ++r) {
    int row = mbase + r;
    if (row < n) H[(size_t)row * HID + ncol] = c[r];
  }
}

// ---------------- edge aggregation, 64 features ----------------

__global__ __launch_bounds__(256) void gcn_agg64(const long long* __restrict__ ei,
                                                 const float* __restrict__ norm,
                                                 const float* __restrict__ Hsrc,
                                                 float* __restrict__ Out, int e, int n) {
  unsigned idx   = blockIdx.x * blockDim.x + threadIdx.x;
  unsigned total = (unsigned)(e + n) * HID;
  if (idx >= total) return;
  int edge = (int)(idx >> 6);
  int f    = (int)(idx & 63u);
  int r, col;
  if (edge < e) {
    r   = (int)ei[edge];
    col = (int)ei[e + edge];
  } else {
    r = col = edge - e;                        // self-loop
  }
  float v = Hsrc[(size_t)r * HID + f] * norm[edge];
  atomicAdd(&Out[(size_t)col * HID + f], v);
}

__global__ __launch_bounds__(256) void gcn_tanh_bias64(float* A,
                                                       const float* __restrict__ b,
                                                       int n) {
  int i = blockIdx.x * blockDim.x + threadIdx.x;
  if (i < n * HID) A[i] = tanhf(A[i] + b[i & 63]);
}

// ---------------- GEMM2: Y = H1(Nx64) @ W2(64x2) (VALU; memory-bound) ----------

__global__ __launch_bounds__(256) void gcn_gemm2(const float* __restrict__ H1,
                                                 const float* __restrict__ W2,
                                                 float* __restrict__ Y, int n) {
  int i = blockIdx.x * blockDim.x + threadIdx.x;
  if (i >= n) return;
  const float* hp = H1 + (size_t)i * HID;
  float a0 = 0.0f, a1 = 0.0f;
#pragma unroll 8
  for (int k = 0; k < HID; ++k) {
    float hv = hp[k];
    a0 = fmaf(hv, W2[2 * k + 0], a0);
    a1 = fmaf(hv, W2[2 * k + 1], a1);
  }
  Y[2 * i + 0] = a0;
  Y[2 * i + 1] = a1;
}

// ---------------- edge aggregation, 2 features ----------------

__global__ __launch_bounds__(256) void gcn_agg2(const long long* __restrict__ ei,
                                                const float* __restrict__ norm,
                                                const float* __restrict__ Y,
                                                float* __restrict__ Out, int e, int n) {
  int i = blockIdx.x * blockDim.x + threadIdx.x;
  if (i >= e + n) return;
  int r, col;
  if (i < e) {
    r   = (int)ei[i];
    col = (int)ei[e + i];
  } else {
    r = col = i - e;
  }
  float nm = norm[i];
  atomicAdd(&Out[2 * col + 0], Y[2 * r + 0] * nm);
  atomicAdd(&Out[2 * col + 1], Y[2 * r + 1] * nm);
}

// ---------------- head: tanh -> linear -> sigmoid ----------------

__global__ __launch_bounds__(256) void gcn_head(const float* __restrict__ A2,
                                                const float* __restrict__ b2,
                                                const float* __restrict__ Wc,
                                                const float* __restrict__ bc,
                                                float* __restrict__ out, int n) {
  int i = blockIdx.x * blockDim.x + threadIdx.x;
  if (i >= n) return;
  float e0 = tanhf(A2[2 * i + 0] + b2[0]);
  float e1 = tanhf(A2[2 * i + 1] + b2[1]);
  float z  = e0 * Wc[0] + e1 * Wc[1] + bc[0];
  out[i]   = 1.0f / (1.0f + expf(-z));
}

// ---------------- launch ----------------

extern "C" void kernel_launch(void* const* d_in, const int* in_sizes, int n_in,
                              void* d_out, int out_size, void* d_ws, size_t ws_size,
                              hipStream_t stream) {
  (void)n_in; (void)out_size; (void)ws_size;
  const float*      x  = (const float*)d_in[0];
  const long long*  ei = (const long long*)d_in[1];   // int64 per reference
  const float*      W1 = (const float*)d_in[2];
  const float*      b1 = (const float*)d_in[3];
  const float*      W2 = (const float*)d_in[4];
  const float*      b2 = (const float*)d_in[5];
  const float*      Wc = (const float*)d_in[6];
  const float*      bc = (const float*)d_in[7];
  float*            out = (float*)d_out;

  const int N = in_sizes[0] / IN_DIM;
  const int E = in_sizes[1] / 2;

  // workspace layout (floats)
  float* ws   = (float*)d_ws;
  float* dis  = ws;                 ws += N;                 // degree -> dis in place
  float* norm = ws;                 ws += (size_t)E + N;
  float* h    = ws;                 ws += (size_t)N * HID;   // GEMM1 output
  float* agg1 = ws;                 ws += (size_t)N * HID;   // layer-1 aggregate -> h1
  float* emb  = ws;                 ws += (size_t)N * 2;     // GEMM2 output
  float* agg2 = ws;                 ws += (size_t)N * 2;     // layer-2 aggregate

  const int T = 256;
  // normalization
  gcn_deg_init <<<(N + T - 1) / T, T, 0, stream>>>(dis, N);
  gcn_deg_count<<<(E + T - 1) / T, T, 0, stream>>>(ei, dis, E);
  gcn_dis      <<<(N + T - 1) / T, T, 0, stream>>>(dis, N);
  gcn_norm     <<<(E + N + T - 1) / T, T, 0, stream>>>(ei, dis, norm, E, N);

  // layer 1
  int mtiles = (N + 15) / 16;
  gcn_gemm1_wmma<<<(mtiles + 1) / 2, T, 0, stream>>>(x, W1, h, N);
  hipMemsetAsync(agg1, 0, (size_t)N * HID * sizeof(float), stream);
  {
    size_t total = (size_t)(E + N) * HID;
    gcn_agg64<<<(unsigned)((total + T - 1) / T), T, 0, stream>>>(ei, norm, h, agg1, E, N);
  }
  gcn_tanh_bias64<<<((size_t)N * HID + T - 1) / T, T, 0, stream>>>(agg1, b1, N);

  // layer 2
  gcn_gemm2<<<(N + T - 1) / T, T, 0, stream>>>(agg1, W2, emb, N);
  hipMemsetAsync(agg2, 0, (size_t)N * 2 * sizeof(float), stream);
  gcn_agg2<<<(E + N + T - 1) / T, T, 0, stream>>>(ei, norm, emb, agg2, E, N);

  // head
  gcn_head<<<(N + T - 1) / T, T, 0, stream>>>(agg2, b2, Wc, bc, out, N);
}